// CQTNSGT_81784767250613
// MI455X (gfx1250) — compile-verified
//
#include <hip/hip_runtime.h>
#include <math.h>
#include <vector>

typedef float v2f __attribute__((ext_vector_type(2)));
typedef float v8f __attribute__((ext_vector_type(8)));

#define TWO_PI_F 6.28318530717958647692f
#define RT 4   // row-tiles (of 16 rows) per wave in the IDFT GEMM

__device__ inline float2 cmulf(float2 a, float2 b) {
    return make_float2(a.x * b.x - a.y * b.y, a.x * b.y + a.y * b.x);
}

// Global load through a uniform base + 32-bit byte offset (GVS addressing).
__device__ inline float4 ld_global_off(const float4* __restrict__ base,
                                       unsigned byteOff) {
    return *(const float4*)((const char*)base + (size_t)byteOff);
}

// ---------------------------------------------------------------------------
// Pack real input rows into interleaved complex (float2) rows.
// ---------------------------------------------------------------------------
__global__ void pack_complex_kernel(const float* __restrict__ x,
                                    float2* __restrict__ out, int N) {
    int row = blockIdx.y;
    int i = blockIdx.x * blockDim.x + threadIdx.x;
    if (i < N) {
        out[(size_t)row * N + i] = make_float2(x[(size_t)row * N + i], 0.0f);
    }
}

// ---------------------------------------------------------------------------
// One Stockham radix-2 stage (autosort, out-of-place ping-pong).
// dir = -1 for forward FFT (jnp.fft.fft convention).
// ---------------------------------------------------------------------------
__global__ void fft_stockham_r2_kernel(const float2* __restrict__ src,
                                       float2* __restrict__ dst,
                                       int N, int Ns, float dir) {
    int row = blockIdx.y;
    int gid = blockIdx.x * blockDim.x + threadIdx.x;
    int half = N >> 1;
    if (gid >= half) return;
    const float2* s = src + (size_t)row * N;
    float2* d = dst + (size_t)row * N;

    int j = gid & (Ns - 1);             // Ns is a power of two
    float ang = dir * TWO_PI_F * (float)j / (float)(2 * Ns);
    float sw, cw;
    __sincosf(ang, &sw, &cw);
    float2 w = make_float2(cw, sw);

    float2 a = s[gid];
    float2 b = cmulf(w, s[gid + half]);
    int od = ((gid - j) << 1) + j;      // (gid/Ns)*2Ns + j
    d[od]      = make_float2(a.x + b.x, a.y + b.y);
    d[od + Ns] = make_float2(a.x - b.x, a.y - b.y);
}

// ---------------------------------------------------------------------------
// Gather spectrum samples, apply windows, and store in the WMMA-fragment
// packed layout: for each even K pair, float4{re[k], re[k+1], im[k], im[k+1]}
// so one b128 load yields the A-fragment pairs {are, aim} with no shuffles.
// Zero-padded to rowsP x Mp so the GEMM loads unconditionally.
// ---------------------------------------------------------------------------
__global__ void gather_window_pack_kernel(const float2* __restrict__ Xf,
                                          const float* __restrict__ giis,
                                          const int* __restrict__ ix,
                                          float4* __restrict__ T,
                                          int nb, int maxM, int Mp,
                                          int rows, int rowsP, int LS) {
    long long idx = (long long)blockIdx.x * blockDim.x + threadIdx.x;
    int Mp2 = Mp >> 1;
    long long total = (long long)rowsP * Mp2;
    if (idx >= total) return;
    int kp = (int)(idx % Mp2);
    int row = (int)(idx / Mp2);
    int k0 = kp << 1;
    int k1 = k0 + 1;
    float4 v = make_float4(0.f, 0.f, 0.f, 0.f);
    if (row < rows) {
        int band = row % nb;
        int batch = row / nb;
        const float* grow = giis + (size_t)band * maxM;
        const int* irow = ix + (size_t)band * maxM;
        const float2* xrow = Xf + (size_t)batch * LS;
        if (k0 < maxM) {
            float2 s = xrow[irow[k0]];
            float g = grow[k0];
            v.x = s.x * g;      // re[k0]
            v.z = s.y * g;      // im[k0]
        }
        if (k1 < maxM) {
            float2 s = xrow[irow[k1]];
            float g = grow[k1];
            v.y = s.x * g;      // re[k1]
            v.w = s.y * g;      // im[k1]
        }
    }
    T[idx] = v;
}

// ---------------------------------------------------------------------------
// Direct inverse DFT as complex GEMM on the WMMA pipe:
//   C[row, n] = (1/M) * sum_k T[row, k] * exp(+2*pi*i*n*k/M)
// One wave32 handles RT 16-row tiles x one 16-col tile; K swept in groups of
// 16 (4 chunks of WMMA-K=4).  DFT fragments are generated on the fly with an
// incremental integer phase recurrence: p(k+4) = (p(k) + 4n) mod M (no int
// mul/div in the loop).  A-loads use uniform-base + 32-bit voffset GLOBAL
// addressing with constant chunk IOFFSETs, and are software-pipelined one
// chunk ahead (double buffered) so the loadcnt wait lands a full chunk of
// WMMAs + sincos after issue.  T is zero-padded and fragment-packed; EXEC
// stays all-ones throughout the loop (WMMA requirement).
// Layouts per ISA 7.12.2:
//   A (16x4 f32): lane = M row; VGPR0 = K {0|2}, VGPR1 = K {1|3} (hi lanes +2)
//   B mirrored with lane = N column.
//   C/D: VGPR v holds M = v + 8*(lane>=16), N = lane&15.
// ---------------------------------------------------------------------------
__global__ void __launch_bounds__(32)
idft_wmma_kernel(const float4* __restrict__ T, float2* __restrict__ out,
                 int rows, int M, int Mp) {
    int tn = blockIdx.x;            // n-tile
    int rg = blockIdx.y;            // group of RT row-tiles
    int lane = threadIdx.x;         // 0..31 (wave32)
    int half = lane >> 4;           // 0 / 1
    int l15 = lane & 15;
    int n0 = tn << 4;
    int r0 = rg * (RT * 16);

    v8f cre[RT], cim[RT];
#pragma unroll
    for (int rt = 0; rt < RT; ++rt) {
        v8f z = {0.f, 0.f, 0.f, 0.f, 0.f, 0.f, 0.f, 0.f};
        cre[rt] = z;
        cim[rt] = z;
    }

    int ncol = n0 + l15;
    int nmod = (ncol < M) ? ncol : 0;       // padded cols: phase irrelevant
    int bk = half << 1;                     // K sub-offset within each 4-chunk
    // running phases p = (k * nmod) mod M for k = bk and k = bk+1
    int pA = (int)(((long long)bk * nmod) % M);
    int pB = (int)(((long long)(bk + 1) * nmod) % M);
    int step4 = (int)(((long long)4 * nmod) % M);
    float invM = 1.0f / (float)M;
    float ph2pi = TWO_PI_F * invM;

    // 32-bit byte offsets into T (T <= ~170MB < 2^31); half folded in so
    // chunk loads use constant instruction offsets (kk * 32 bytes).
    int Mp2 = Mp >> 1;
    unsigned off[RT];
#pragma unroll
    for (int rt = 0; rt < RT; ++rt) {
        off[rt] = (unsigned)((((size_t)(r0 + rt * 16 + l15) * Mp2) + half) * 16u);
    }

    // prologue: preload chunk 0
    float4 tc[RT];
#pragma unroll
    for (int rt = 0; rt < RT; ++rt) {
        tc[rt] = ld_global_off(T, off[rt]);
    }

    int kTiles = Mp >> 4;                   // Mp is a multiple of 16
    for (int kt = 0; kt < kTiles; ++kt) {
#pragma unroll
        for (int kk = 0; kk < 4; ++kk) {
            // ---- prefetch next chunk into the other buffer ----
            // (kk<3: next chunk of this group at +32B steps; kk==3: first
            //  chunk of the next group at +128B.  Tail prefetch over-reads
            //  <=128B past T; workspace reserves slack.)
            float4 tnx[RT];
            unsigned nextOff = (kk < 3) ? (unsigned)((kk + 1) * 32)
                                        : (unsigned)128;
#pragma unroll
            for (int rt = 0; rt < RT; ++rt) {
                tnx[rt] = ld_global_off(T, off[rt] + nextOff);
            }
            // ---- B fragment: exp(+2*pi*i*k*n/M) via incremental phase ----
            float a0 = ph2pi * (float)pA;
            float a1 = ph2pi * (float)pB;
            float s0, c0, s1, c1;
            __sincosf(a0, &s0, &c0);
            __sincosf(a1, &s1, &c1);
            pA += step4; if (pA >= M) pA -= M;
            pB += step4; if (pB >= M) pB -= M;
            v2f bcos  = { c0,  c1};
            v2f bsin  = { s0,  s1};
            v2f bnsin = {-s0, -s1};
            // ---- complex MAC via 4 real WMMAs per row-tile (uses tc) ----
#pragma unroll
            for (int rt = 0; rt < RT; ++rt) {
                v2f are = {tc[rt].x, tc[rt].y};
                v2f aim = {tc[rt].z, tc[rt].w};
                cre[rt] = __builtin_amdgcn_wmma_f32_16x16x4_f32(
                    false, are, false, bcos, (short)0, cre[rt], false, false);
                cre[rt] = __builtin_amdgcn_wmma_f32_16x16x4_f32(
                    false, aim, false, bnsin, (short)0, cre[rt], false, false);
                cim[rt] = __builtin_amdgcn_wmma_f32_16x16x4_f32(
                    false, are, false, bsin, (short)0, cim[rt], false, false);
                cim[rt] = __builtin_amdgcn_wmma_f32_16x16x4_f32(
                    false, aim, false, bcos, (short)0, cim[rt], false, false);
            }
            // ---- rotate double buffer ----
#pragma unroll
            for (int rt = 0; rt < RT; ++rt) {
                tc[rt] = tnx[rt];
            }
        }
#pragma unroll
        for (int rt = 0; rt < RT; ++rt) {
            off[rt] += 128u;                // advance one 16-K group
        }
    }

    if (ncol < M) {
#pragma unroll
        for (int rt = 0; rt < RT; ++rt) {
            int rbase = r0 + rt * 16 + (half << 3);
#pragma unroll
            for (int v = 0; v < 8; ++v) {
                int orow = rbase + v;
                if (orow < rows) {
                    out[(size_t)orow * M + ncol] =
                        make_float2(cre[rt][v] * invM, cim[rt][v] * invM);
                }
            }
        }
    }
}

// ---------------------------------------------------------------------------
// Host-side replica of the numpy filterbank sizing (deterministic, cheap).
// ---------------------------------------------------------------------------
static void compute_dims(int* nb_out, int* maxM_out) {
    const double SRd = 44100.0;
    const double LSd = 524288.0;
    const double FMIN = 32.7, FMAX = 21500.0;
    const int BPO = 48;
    const double nf = SRd / 2.0;
    const double Q = 1.0 / (pow(2.0, 1.0 / BPO) - pow(2.0, -1.0 / BPO));

    int b = (int)floor(BPO * log2(FMAX / FMIN)) + 1;
    std::vector<double> f;
    for (int k = 0; k < b; ++k) {
        double fk = FMIN * pow(2.0, (double)k / BPO);
        if (fk > 0.0 && fk < nf) f.push_back(fk);
    }
    int lbas = (int)f.size();

    std::vector<double> frqs;
    frqs.push_back(0.0);
    for (double v : f) frqs.push_back(v);
    frqs.push_back(nf);

    std::vector<double> fbas = frqs;
    for (int i = (int)frqs.size() - 2; i >= 1; --i) fbas.push_back(SRd - frqs[i]);
    for (double& v : fbas) v *= LSd / SRd;

    int n = (int)fbas.size();                 // = 2*lbas + 2
    std::vector<long long> M(n, 0);
    M[0] = llround(2.0 * fbas[1]);
    M[1] = llround(fbas[1] / Q);
    for (int k = 2; k <= lbas; ++k) M[k] = llround(fbas[k + 1] - fbas[k - 1]);
    M[lbas + 1] = llround(fbas[lbas] / Q);
    for (int j = 0; j < lbas && (lbas + 2 + j) < n; ++j)
        M[lbas + 2 + j] = M[lbas - j];

    long long mx = 4;
    for (int k = 0; k < n; ++k) {
        if (M[k] < 4) M[k] = 4;
        if (M[k] > mx) mx = M[k];
    }
    *nb_out = n;
    *maxM_out = (int)mx;
}

// ---------------------------------------------------------------------------
extern "C" void kernel_launch(void* const* d_in, const int* in_sizes, int n_in,
                              void* d_out, int out_size, void* d_ws,
                              size_t ws_size, hipStream_t stream) {
    const float* x    = (const float*)d_in[0];   // [2, 524288] f32
    const float* giis = (const float*)d_in[1];   // [nb, maxM] f32
    const int*   ix   = (const int*)d_in[2];     // [nb, maxM] i32

    const int N = 524288;
    const int B = 2;

    int nb = 0, maxM = 0;
    compute_dims(&nb, &maxM);
    if (nb > 0 && (long long)nb * maxM != (long long)in_sizes[1]) {
        // keep product consistent with the actual input if the replica drifts
        maxM = in_sizes[1] / nb;
    }

    const int rows = B * nb;
    const int Mp = (maxM + 15) & ~15;                   // col-padded K extent
    const int rowGroups = (rows + RT * 16 - 1) / (RT * 16);
    const int rowsP = rowGroups * RT * 16;              // row-padded extent

    // Workspace layout: two FFT ping-pong buffers + packed/padded T matrix
    // (+128B tail slack for the pipelined prefetch over-read).
    float2* bufA = (float2*)d_ws;
    float2* bufB = bufA + (size_t)B * N;
    float4* T    = (float4*)(bufB + (size_t)B * N);

    // 1) pack real input to complex
    {
        dim3 blk(256);
        dim3 grd((N + 255) / 256, B);
        pack_complex_kernel<<<grd, blk, 0, stream>>>(x, bufA, N);
    }

    // 2) forward FFT (19 Stockham radix-2 stages, dir = -1)
    float2* src = bufA;
    float2* dst = bufB;
    {
        dim3 blk(256);
        dim3 grd((N / 2 + 255) / 256, B);
        for (int Ns = 1; Ns < N; Ns <<= 1) {
            fft_stockham_r2_kernel<<<grd, blk, 0, stream>>>(src, dst, N, Ns, -1.0f);
            float2* t = src; src = dst; dst = t;
        }
    }
    // spectrum now in `src`

    // 3) gather + window into fragment-packed, zero-padded T
    {
        long long total = (long long)rowsP * (Mp >> 1);
        int blocks = (int)((total + 255) / 256);
        gather_window_pack_kernel<<<blocks, 256, 0, stream>>>(
            src, giis, ix, T, nb, maxM, Mp, rows, rowsP, N);
    }

    // 4) inverse DFT as WMMA GEMM, writing [batch, band, n, {re,im}] directly
    {
        dim3 grd(Mp / 16, rowGroups);
        idft_wmma_kernel<<<grd, 32, 0, stream>>>(T, (float2*)d_out, rows, maxM, Mp);
    }
}